// CausalSelfAttention_20263655703354
// MI455X (gfx1250) — compile-verified
//
#include <hip/hip_runtime.h>
#include <math.h>
#include <stdint.h>

// ---------------------------------------------------------------------------
// CoPE causal self-attention, fp32, CDNA5 (gfx1250, wave32, WMMA).
// B=1, T=2048, E=768, H=12, HD=64, NPOS=64, TEMP=1.
//
// Op is bandwidth-bound (~25 GFLOP vs ~80 MB HBM traffic @ 23.3 TB/s), so we
// keep full fp32 precision and use V_WMMA_F32_16X16X4_F32 for every matmul.
// The O(T^2) score strip lives in the 320KB WGP LDS (~148KB/WG incl. K/V
// staging buffers) because the CoPE suffix-scan needs whole rows before
// softmax (no flash-attention possible).
//
// CDNA5-specific paths used:
//   * v_wmma_f32_16x16x4_f32 for QKV GEMM, q@cope_emb, QK^T, P@V, out GEMM
//   * global_load_async_to_lds_b128 + s_wait_asynccnt double-buffered K/V
//     tile staging (async engine overlaps with matrix pipe)
//   * global_prefetch_b8 on the GEMM weight streams
// ---------------------------------------------------------------------------

typedef float v2f __attribute__((ext_vector_type(2)));
typedef float v8f __attribute__((ext_vector_type(8)));

constexpr int kT    = 2048;
constexpr int kE    = 768;
constexpr int kH    = 12;
constexpr int kHD   = 64;
constexpr int kNPOS = 64;
constexpr int kQB   = 16;          // queries per workgroup
constexpr int kNQB  = kT / kQB;    // 128 query blocks
constexpr int kSP   = 2052;        // padded LDS score-row stride (floats):
                                   // 2052 % 64 == 4 -> 16 scanning lanes hit
                                   // 16 distinct banks (stride 4 banks/lane)

static __device__ __forceinline__ v8f wmma_f32(v2f a, v2f b, v8f c) {
  // 8-arg form: (neg_a, A, neg_b, B, c_mod, C, reuse_a, reuse_b)
  return __builtin_amdgcn_wmma_f32_16x16x4_f32(false, a, false, b,
                                               (short)0, c, false, false);
}

// Async-copy one 16x64 fp32 tile (4KB) global -> LDS.
// 32 lanes x 8 x b128 = 4KB; tracked on ASYNCcnt (8 outstanding per call).
static __device__ __forceinline__ void async_copy_tile(float* dst_lds,
                                                       const float* src,
                                                       int lane) {
  uint32_t lds = (uint32_t)(uintptr_t)dst_lds + (uint32_t)lane * 16u;
  const char* g = (const char*)src + lane * 16;
#pragma unroll
  for (int i = 0; i < 8; ++i) {
    asm volatile("global_load_async_to_lds_b128 %0, %1, off"
                 :: "v"(lds + (uint32_t)i * 512u), "v"(g + i * 512)
                 : "memory");
  }
}
static __device__ __forceinline__ void wait_async_le8() {
  asm volatile("s_wait_asynccnt 0x8" ::: "memory");
}
static __device__ __forceinline__ void wait_async_0() {
  asm volatile("s_wait_asynccnt 0x0" ::: "memory");
}

// ---------------------------------------------------------------------------
// Kernel 1: qkv = x @ w_attn, scattered into q/k/v [H][T][HD] workspace.
// Grid: (T/16, 3E/64). Each wave owns a 16x64 output strip: the A fragment
// is reused across 4 WMMAs per K-step (4x arithmetic intensity on A).
// ---------------------------------------------------------------------------
__global__ __launch_bounds__(32)
void qkv_gemm(const float* __restrict__ x, const float* __restrict__ w,
              float* __restrict__ qws, float* __restrict__ kws,
              float* __restrict__ vws) {
  const int lane = threadIdx.x;
  const int hl   = lane >> 4;   // half-wave: K {0,1} vs {2,3}; M 0-7 vs 8-15
  const int l    = lane & 15;
  const int row0 = blockIdx.x * 16;
  const int col0 = blockIdx.y * 64;

  const v8f zero = {0.f,0.f,0.f,0.f,0.f,0.f,0.f,0.f};
  v8f acc[4] = {zero, zero, zero, zero};

  const float* arow = &x[(row0 + l) * kE];
  for (int k0 = 0; k0 < kE; k0 += 4) {
    const int ka = k0 + 2 * hl;
    v2f a;
    a.x = arow[ka];
    a.y = arow[ka + 1];
    const float* b0 = &w[ka * (3 * kE) + col0 + l];
    const float* b1 = b0 + 3 * kE;
    __builtin_prefetch(b0 + 8 * 3 * kE, 0, 1);   // global_prefetch_b8, 8 rows ahead
#pragma unroll
    for (int j = 0; j < 4; ++j) {
      v2f b;
      b.x = b0[16 * j];
      b.y = b1[16 * j];
      acc[j] = wmma_f32(a, b, acc[j]);
    }
  }
  // scatter output columns into q/k/v [H][T][HD]
#pragma unroll
  for (int j = 0; j < 4; ++j) {
    const int cn     = col0 + 16 * j + l;
    const int sec    = cn / kE;          // 0=q 1=k 2=v
    const int within = cn % kE;
    const int hh     = within / kHD;
    const int d      = within % kHD;
    float* dst = (sec == 0) ? qws : (sec == 1) ? kws : vws;
    for (int i = 0; i < 8; ++i) {
      const int r = row0 + i + 8 * hl;
      dst[(hh * kT + r) * kHD + d] = acc[j][i];
    }
  }
}

// ---------------------------------------------------------------------------
// Kernel 2: per (head, 16-query block): S = QK^T/8 into LDS strip (WMMA, K
// tiles async-staged into a double buffer), CoPE logits q@cope_emb (WMMA),
// per-row backward gate scan + bias + softmax in place, then P@V (WMMA, V
// tiles async-staged). Grid: H*NQB blocks, 1 wave each.
// ---------------------------------------------------------------------------
extern __shared__ float smem[];

__global__ __launch_bounds__(32)
void cope_attn(const float* __restrict__ qws, const float* __restrict__ kws,
               const float* __restrict__ vws, const float* __restrict__ cope,
               float* __restrict__ yws) {
  float* sQ  = smem;                        // 16 x 64
  float* sL  = sQ + kQB * kHD;              // 16 x 64  (interp logits)
  float* sB0 = sL + kQB * kHD;              // 16 x 64  K/V stage buf 0
  float* sB1 = sB0 + kQB * kHD;             // 16 x 64  K/V stage buf 1
  float* sS  = sB1 + kQB * kHD;             // 16 x kSP (score / prob strip)

  const int lane = threadIdx.x;
  const int hl   = lane >> 4;
  const int l    = lane & 15;
  const int head = blockIdx.x / kNQB;
  const int qblk = blockIdx.x % kNQB;
  const int q0   = qblk * kQB;
  const int nkb  = qblk + 1;                // causal: key blocks 0..qblk
  const float scale = 0.125f;               // 1/sqrt(64)
  const v8f zero = {0.f,0.f,0.f,0.f,0.f,0.f,0.f,0.f};

  // ---- stage Q tile ----
  for (int idx = lane; idx < kQB * kHD; idx += 32) {
    const int r = idx / kHD, d = idx % kHD;
    sQ[idx] = qws[(head * kT + q0 + r) * kHD + d];
  }
  __syncthreads();

  // ---- CoPE interpolation logits: L = Q(16x64) @ cope_emb(64x64) ----
  // (overlaps with the first async K-tile fetch issued below)
  async_copy_tile(sB0, &kws[(head * kT) * kHD], lane);  // K block 0
  for (int pt = 0; pt < 4; ++pt) {
    v8f c = zero;
    for (int d0 = 0; d0 < kHD; d0 += 4) {
      const int ka = d0 + 2 * hl;
      v2f a, b;
      a.x = sQ[l * kHD + ka];
      a.y = sQ[l * kHD + ka + 1];
      b.x = cope[ka * kNPOS + pt * 16 + l];
      b.y = cope[(ka + 1) * kNPOS + pt * 16 + l];
      c = wmma_f32(a, b, c);
    }
    for (int i = 0; i < 8; ++i)
      sL[(i + 8 * hl) * kNPOS + pt * 16 + l] = c[i];
  }

  // ---- scores: S[16][key] = Q @ K^T * scale, K tiles double-buffered ----
  for (int kb = 0; kb < nkb; ++kb) {
    float* cur = (kb & 1) ? sB1 : sB0;
    float* nxt = (kb & 1) ? sB0 : sB1;
    if (kb + 1 < nkb) {
      async_copy_tile(nxt, &kws[(head * kT + (kb + 1) * kQB) * kHD], lane);
      wait_async_le8();   // in-order completion => cur tile resident
    } else {
      wait_async_0();
    }
    const int kbase = kb * kQB;
    v8f c = zero;
    for (int d0 = 0; d0 < kHD; d0 += 4) {
      const int ka = d0 + 2 * hl;
      v2f a, b;
      a.x = sQ[l * kHD + ka];
      a.y = sQ[l * kHD + ka + 1];
      b.x = cur[l * kHD + ka];       // B[d][n] = K[kbase+n][d] from LDS
      b.y = cur[l * kHD + ka + 1];
      c = wmma_f32(a, b, c);
    }
    for (int i = 0; i < 8; ++i)
      sS[(i + 8 * hl) * kSP + kbase + l] = c[i] * scale;
  }
  __syncthreads();

  // ---- CoPE backward scan + bias + softmax, one row per lane (0..15) ----
  // pos[k] = min(sum_{j=k..q} sigmoid(S[j]), 63); inherently serial in k.
  if (lane < kQB) {
    const int row = lane;
    const int qg  = q0 + row;                 // global query index
    float* s = &sS[row * kSP];
    float run = 0.f, mx = -3.4e38f;
    for (int j = qg; j >= 0; --j) {           // suffix scan, high -> low
      const float sv = s[j];
      const float g  = 1.0f / (1.0f + __expf(-sv));
      run += g;
      const float pos = fminf(run, (float)(kNPOS - 1));
      const float pf  = floorf(pos);
      const int ipf = (int)pf;
      const int ipc = (pos > pf && ipf < kNPOS - 1) ? ipf + 1 : ipf;  // ceil
      const float w = pos - pf;
      const float aw = sv + sL[row * kNPOS + ipc] * w
                          + sL[row * kNPOS + ipf] * (1.0f - w);
      s[j] = aw;
      mx = fmaxf(mx, aw);
    }
    float sum = 0.f;
    for (int j = 0; j <= qg; ++j) {
      const float e = __expf(s[j] - mx);      // TEMP = 1
      s[j] = e;
      sum += e;
    }
    const float inv = 1.0f / sum;
    for (int j = 0; j <= qg; ++j) s[j] *= inv;
    const int rowlen = nkb * kQB;             // zero causal tail for P@V
    for (int j = qg + 1; j < rowlen; ++j) s[j] = 0.f;
  }
  __syncthreads();

  // ---- y = P(16 x 16*nkb) @ V -> 16x64, V tiles double-buffered ----
  v8f acc[4] = {zero, zero, zero, zero};
  async_copy_tile(sB0, &vws[(head * kT) * kHD], lane);  // V block 0
  for (int kb = 0; kb < nkb; ++kb) {
    float* cur = (kb & 1) ? sB1 : sB0;
    float* nxt = (kb & 1) ? sB0 : sB1;
    if (kb + 1 < nkb) {
      async_copy_tile(nxt, &vws[(head * kT + (kb + 1) * kQB) * kHD], lane);
      wait_async_le8();
    } else {
      wait_async_0();
    }
    const int kbase = kb * kQB;
#pragma unroll
    for (int c4 = 0; c4 < 4; ++c4) {
      const int klocal = c4 * 4 + 2 * hl;      // key within the 16-key block
      v2f a;
      a.x = sS[l * kSP + kbase + klocal];
      a.y = sS[l * kSP + kbase + klocal + 1];
#pragma unroll
      for (int dt = 0; dt < 4; ++dt) {
        v2f b;
        b.x = cur[klocal * kHD + dt * 16 + l];
        b.y = cur[(klocal + 1) * kHD + dt * 16 + l];
        acc[dt] = wmma_f32(a, b, acc[dt]);
      }
    }
  }
  for (int dt = 0; dt < 4; ++dt)
    for (int i = 0; i < 8; ++i)
      yws[(head * kT + q0 + i + 8 * hl) * kHD + dt * 16 + l] = acc[dt][i];
}

// ---------------------------------------------------------------------------
// Kernel 3: out = y @ w_proj, gathering y from [H][T][HD] on the fly.
// Grid: (T/16, E/64), 16x64 output strip per wave (A fragment reused 4x).
// ---------------------------------------------------------------------------
__global__ __launch_bounds__(32)
void proj_gemm(const float* __restrict__ yws, const float* __restrict__ w,
               float* __restrict__ out) {
  const int lane = threadIdx.x;
  const int hl   = lane >> 4;
  const int l    = lane & 15;
  const int row0 = blockIdx.x * 16;
  const int col0 = blockIdx.y * 64;

  const v8f zero = {0.f,0.f,0.f,0.f,0.f,0.f,0.f,0.f};
  v8f acc[4] = {zero, zero, zero, zero};
  for (int k0 = 0; k0 < kE; k0 += 4) {
    const int ka = k0 + 2 * hl;        // even, so ka and ka+1 share a head
    const int hh = ka / kHD, d = ka % kHD;
    v2f a;
    a.x = yws[(hh * kT + row0 + l) * kHD + d];
    a.y = yws[(hh * kT + row0 + l) * kHD + d + 1];
    const float* b0 = &w[ka * kE + col0 + l];
    const float* b1 = b0 + kE;
    __builtin_prefetch(b0 + 8 * kE, 0, 1);
#pragma unroll
    for (int j = 0; j < 4; ++j) {
      v2f b;
      b.x = b0[16 * j];
      b.y = b1[16 * j];
      acc[j] = wmma_f32(a, b, acc[j]);
    }
  }
#pragma unroll
  for (int j = 0; j < 4; ++j)
    for (int i = 0; i < 8; ++i)
      out[(row0 + i + 8 * hl) * kE + col0 + 16 * j + l] = acc[j][i];
}

// ---------------------------------------------------------------------------
extern "C" void kernel_launch(void* const* d_in, const int* in_sizes, int n_in,
                              void* d_out, int out_size, void* d_ws,
                              size_t ws_size, hipStream_t stream) {
  const float* x      = (const float*)d_in[0];
  const float* w_attn = (const float*)d_in[1];
  const float* w_proj = (const float*)d_in[2];
  const float* cope   = (const float*)d_in[3];

  float* ws = (float*)d_ws;
  const size_t per = (size_t)kH * kT * kHD;   // 1.5M floats per tensor
  float* qws = ws;
  float* kws = ws + per;
  float* vws = ws + 2 * per;
  float* yws = ws + 3 * per;

  // 1) QKV projection (fp32 WMMA GEMM, scatter to per-head layout)
  dim3 g1(kT / 16, 3 * kE / 64);
  qkv_gemm<<<g1, 32, 0, stream>>>(x, w_attn, qws, kws, vws);

  // 2) CoPE attention: score strip + Q/L tiles + K/V double buffer in LDS
  const size_t smem_bytes =
      (size_t)(4 * kQB * kHD + kQB * kSP) * sizeof(float);  // ~147.6 KB
  cope_attn<<<dim3(kH * kNQB), 32, smem_bytes, stream>>>(qws, kws, vws, cope,
                                                         yws);

  // 3) Output projection
  dim3 g3(kT / 16, kE / 64);
  proj_gemm<<<g3, 32, 0, stream>>>(yws, w_proj, (float*)d_out);
}